// CellGen_730144440866
// MI455X (gfx1250) — compile-verified
//
#include <hip/hip_runtime.h>
#include <hip/hip_bf16.h>

// Problem constants (from reference): B=4, S=2048, D=1024, E=8, H=64, top-2.
#define T_TOK 8192
#define DIM   1024
#define HID   64
#define NEXP  8
#define SEQ   2048
#define NB    4
#define GP    68   // LDS pitch for G tile (68 % 64 == 4 -> conflict-free WMMA A-frag reads)

typedef float v2f __attribute__((ext_vector_type(2)));
typedef float v8f __attribute__((ext_vector_type(8)));

// D = A(16x4) * B(4x16) + C, all fp32.  8-arg form per probe:
// (neg_a, A, neg_b, B, c_mod, C, reuse_a, reuse_b)
static __device__ __forceinline__ v8f wmma4(v2f a, v2f b, v8f c) {
  return __builtin_amdgcn_wmma_f32_16x16x4_f32(false, a, false, b, (short)0, c,
                                               false, false);
}

// silu via v_exp_f32 + v_rcp_f32 (avoids the IEEE div_scale/div_fixup chain)
static __device__ __forceinline__ float silu(float v) {
  return v * __builtin_amdgcn_rcpf(1.0f + __expf(-v));
}

// ---------------------------------------------------------------------------
// Kernel 1: gate logits + fp32 softmax + top-2 renorm -> dense routing weights
// One wave per token; Wg staged in LDS.
// ---------------------------------------------------------------------------
__global__ __launch_bounds__(256) void moe_gate_kernel(
    const float* __restrict__ x, const float* __restrict__ Wg,
    float* __restrict__ gate_out, float* __restrict__ rw_out,
    float* __restrict__ wden) {
  __shared__ float wg_s[NEXP * DIM];  // 32 KB
  const int tid = threadIdx.x;
  for (int i = tid; i < NEXP * DIM; i += 256) wg_s[i] = Wg[i];
  __syncthreads();

  const int wave = tid >> 5;
  const int lane = tid & 31;
  const int t = blockIdx.x * 8 + wave;  // grid = T/8 blocks

  float p[NEXP];
#pragma unroll
  for (int e = 0; e < NEXP; ++e) p[e] = 0.0f;

  for (int d = lane; d < DIM; d += 32) {
    const float xv = x[(size_t)t * DIM + d];
#pragma unroll
    for (int e = 0; e < NEXP; ++e) p[e] += xv * wg_s[e * DIM + d];
  }
#pragma unroll
  for (int e = 0; e < NEXP; ++e) {
#pragma unroll
    for (int off = 16; off >= 1; off >>= 1)
      p[e] += __shfl_xor(p[e], off, 32);
  }

  if (lane == 0) {
    float mx = p[0];
#pragma unroll
    for (int e = 1; e < NEXP; ++e) mx = fmaxf(mx, p[e]);
    float r[NEXP];
    float s = 0.0f;
#pragma unroll
    for (int e = 0; e < NEXP; ++e) { r[e] = __expf(p[e] - mx); s += r[e]; }
    const float inv = 1.0f / s;
#pragma unroll
    for (int e = 0; e < NEXP; ++e) {
      r[e] *= inv;
      gate_out[(size_t)t * NEXP + e] = p[e];
      rw_out[(size_t)t * NEXP + e]   = r[e];
    }
    // top-2, first-index-wins tie order (matches jax.lax.top_k)
    int s0 = 0;
#pragma unroll
    for (int e = 1; e < NEXP; ++e) if (r[e] > r[s0]) s0 = e;
    int s1 = (s0 == 0) ? 1 : 0;
#pragma unroll
    for (int e = 0; e < NEXP; ++e) if (e != s1 && e != s0 && r[e] > r[s1]) s1 = e;
    const float iw = 1.0f / (r[s0] + r[s1]);
#pragma unroll
    for (int e = 0; e < NEXP; ++e) wden[(size_t)t * NEXP + e] = 0.0f;
    wden[(size_t)t * NEXP + s0] = r[s0] * iw;
    wden[(size_t)t * NEXP + s1] = r[s1] * iw;
  }
}

// ---------------------------------------------------------------------------
// Kernel 2: dense SwiGLU over all experts via f32 WMMA, routing weight folded
// into G rows so GEMM2 accumulates across experts in the WMMA C operand.
// Workgroup = 4 waves = 16 tokens x 1024 out cols. 512 workgroups.
// ---------------------------------------------------------------------------
__global__ __launch_bounds__(128) void moe_ffn_kernel(
    const float* __restrict__ x, const float* __restrict__ W1,
    const float* __restrict__ W2, const float* __restrict__ W3,
    const float* __restrict__ wden, float* __restrict__ outF) {
  __shared__ float G[16 * GP];        // scaled silu(h1)*h3, 16 tokens x 64
  __shared__ float wd[16 * NEXP];     // routing weights for this token tile

  const int t0   = blockIdx.x * 16;
  const int tid  = threadIdx.x;
  const int wave = tid >> 5;
  const int lane = tid & 31;
  const int lo   = lane & 15;   // M (A,C rows) or N (B,C cols)
  const int hi   = lane >> 4;   // K half select

  for (int i = tid; i < 16 * NEXP; i += 128) wd[i] = wden[(size_t)t0 * NEXP + i];
  __syncthreads();

  // 16 output n-tiles per wave (wave owns cols [wave*256, wave*256+256))
  v8f accO[16];
#pragma unroll
  for (int i = 0; i < 16; ++i)
    accO[i] = (v8f){0.f, 0.f, 0.f, 0.f, 0.f, 0.f, 0.f, 0.f};

  const int ncol0 = wave * 16;  // this wave's H-tile in phase A

#pragma unroll 1
  for (int e = 0; e < NEXP; ++e) {
    const float* __restrict__ W1e = W1 + (size_t)e * DIM * HID;
    const float* __restrict__ W3e = W3 + (size_t)e * DIM * HID;
    const float* __restrict__ W2e = W2 + (size_t)e * HID * DIM;

    // ---- Phase A: h1 = x@W1[e], h3 = x@W3[e] for H cols [ncol0, ncol0+16)
    v8f a1 = (v8f){0.f, 0.f, 0.f, 0.f, 0.f, 0.f, 0.f, 0.f};
    v8f a3 = (v8f){0.f, 0.f, 0.f, 0.f, 0.f, 0.f, 0.f, 0.f};
#pragma unroll 4
    for (int k0 = 0; k0 < DIM; k0 += 4) {
      const int ka = k0 + hi * 2;
      v2f af, b1f, b3f;
      // A-frag: lane holds row m=lo, K = ka, ka+1 (L2-resident x)
      af[0] = x[(size_t)(t0 + lo) * DIM + ka];
      af[1] = x[(size_t)(t0 + lo) * DIM + ka + 1];
      // B-frag: lane holds col n=ncol0+lo, K = ka, ka+1
      b1f[0] = W1e[(size_t)ka * HID + ncol0 + lo];
      b1f[1] = W1e[(size_t)(ka + 1) * HID + ncol0 + lo];
      b3f[0] = W3e[(size_t)ka * HID + ncol0 + lo];
      b3f[1] = W3e[(size_t)(ka + 1) * HID + ncol0 + lo];
      a1 = wmma4(af, b1f, a1);
      a3 = wmma4(af, b3f, a3);
    }

    // ---- G = silu(h1) * h3, scaled per-row by routing weight, to LDS
#pragma unroll
    for (int j = 0; j < 8; ++j) {
      const int m = j + 8 * hi;  // C/D layout: VGPR j -> M=j (lanes0-15)/j+8
      float gv = silu(a1[j]) * a3[j];
      gv *= wd[m * NEXP + e];
      G[m * GP + ncol0 + lo] = gv;
    }
    __syncthreads();

    // ---- Phase B: out += Gs @ W2[e], accumulate across experts in accO
#pragma unroll
    for (int i = 0; i < 16; ++i) {
      const int nc = (wave * 16 + i) * 16 + lo;  // output column
      v8f acc = accO[i];
#pragma unroll
      for (int k0 = 0; k0 < HID; k0 += 4) {
        const int ka = k0 + hi * 2;
        v2f af, bf;
        af[0] = G[lo * GP + ka];
        af[1] = G[lo * GP + ka + 1];
        bf[0] = W2e[(size_t)ka * DIM + nc];
        bf[1] = W2e[(size_t)(ka + 1) * DIM + nc];
        acc = wmma4(af, bf, acc);
      }
      accO[i] = acc;
    }
    __syncthreads();  // G reused next expert
  }

  // ---- Write final hidden states
#pragma unroll
  for (int i = 0; i < 16; ++i) {
    const int nc = (wave * 16 + i) * 16 + lo;
#pragma unroll
    for (int j = 0; j < 8; ++j) {
      const int m = j + 8 * hi;
      outF[(size_t)(t0 + m) * DIM + nc] = accO[i][j];
    }
  }
}

// ---------------------------------------------------------------------------
// Kernel 3: per-expert cumulative classifier logits on token s=0 of each batch.
// p[e][b] = wden[t][e] * ( (silu(x W1[e]) * (x W3[e])) . (W2[e] @ Wc) )
// expert_logits[e][b] = cumsum_e p + bc.  One wave per expert.
// ---------------------------------------------------------------------------
__global__ __launch_bounds__(256) void moe_cls_kernel(
    const float* __restrict__ x, const float* __restrict__ W1,
    const float* __restrict__ W2, const float* __restrict__ W3,
    const float* __restrict__ Wc, const float* __restrict__ bc,
    const float* __restrict__ wden, float* __restrict__ elog) {
  __shared__ float pmat[NEXP * NB];
  const int tid = threadIdx.x;
  const int e = tid >> 5;
  const int lane = tid & 31;
  const int hA = lane, hB = lane + 32;

  const float* W2e = W2 + (size_t)e * HID * DIM;
  float vA = 0.f, vB = 0.f;
  for (int d = 0; d < DIM; ++d) {
    const float wc = Wc[d];
    vA += W2e[(size_t)hA * DIM + d] * wc;
    vB += W2e[(size_t)hB * DIM + d] * wc;
  }

  const float* W1e = W1 + (size_t)e * DIM * HID;
  const float* W3e = W3 + (size_t)e * DIM * HID;
  for (int b = 0; b < NB; ++b) {
    const int t = b * SEQ;  // token (b, s=0)
    float h1A = 0.f, h1B = 0.f, h3A = 0.f, h3B = 0.f;
    for (int d = 0; d < DIM; ++d) {
      const float xv = x[(size_t)t * DIM + d];
      h1A += xv * W1e[(size_t)d * HID + hA];
      h1B += xv * W1e[(size_t)d * HID + hB];
      h3A += xv * W3e[(size_t)d * HID + hA];
      h3B += xv * W3e[(size_t)d * HID + hB];
    }
    float part = silu(h1A) * h3A * vA + silu(h1B) * h3B * vB;
#pragma unroll
    for (int off = 16; off >= 1; off >>= 1)
      part += __shfl_xor(part, off, 32);
    if (lane == 0) pmat[e * NB + b] = wden[(size_t)t * NEXP + e] * part;
  }
  __syncthreads();
  if (tid < NB) {
    float run = 0.f;
    for (int e2 = 0; e2 < NEXP; ++e2) {
      run += pmat[e2 * NB + tid];
      elog[e2 * NB + tid] = run + bc[0];
    }
  }
}

extern "C" void kernel_launch(void* const* d_in, const int* in_sizes, int n_in,
                              void* d_out, int out_size, void* d_ws,
                              size_t ws_size, hipStream_t stream) {
  const float* x  = (const float*)d_in[0];
  // d_in[1] = tgt_pad (bool, unused by this path)
  const float* Wg = (const float*)d_in[2];
  const float* W1 = (const float*)d_in[3];
  const float* W2 = (const float*)d_in[4];
  const float* W3 = (const float*)d_in[5];
  const float* Wc = (const float*)d_in[6];
  const float* bc = (const float*)d_in[7];

  float* out = (float*)d_out;
  float* outF = out;                                   // [T, D]
  float* gate = outF + (size_t)T_TOK * DIM;            // [T, E]
  float* elog = gate + (size_t)T_TOK * NEXP;           // [E, B, 1]
  float* rw   = elog + (size_t)NEXP * NB;              // [T, E]

  float* wden = (float*)d_ws;                          // [T, E] dense routing

  moe_gate_kernel<<<T_TOK / 8, 256, 0, stream>>>(x, Wg, gate, rw, wden);
  moe_ffn_kernel<<<T_TOK / 16, 128, 0, stream>>>(x, W1, W2, W3, wden, outF);
  moe_cls_kernel<<<1, 256, 0, stream>>>(x, W1, W2, W3, Wc, bc, wden, elog);
}